// BayesianGNN_76149770158503
// MI455X (gfx1250) — compile-verified
//
#include <hip/hip_runtime.h>
#include <hip/hip_bf16.h>

typedef __attribute__((ext_vector_type(2))) float v2f;
typedef __attribute__((ext_vector_type(8))) float v8f;

#define N_NODES 50000
#define E_EDGES 800000
#define IN_F    96
#define HID_F   128
#define OUT_F   64
#define T_S     8
#define MWAVES  512   // waves per N-tile column group in the GEMMs

// ---------------------------------------------------------------- zero fill
__global__ __launch_bounds__(256) void zero_f32(float* __restrict__ p, int n4) {
    int i = blockIdx.x * 256 + threadIdx.x;
    if (i < n4) {
        float4 z = make_float4(0.f, 0.f, 0.f, 0.f);
        *(float4*)(p + (size_t)i * 4) = z;
    }
}

// ------------------------------------------------- masked weights (tiny op)
__global__ __launch_bounds__(256) void mask_weights(const float* __restrict__ W1,
                                                    const float* __restrict__ m1,
                                                    const float* __restrict__ W2,
                                                    const float* __restrict__ m2,
                                                    float* __restrict__ W1m,
                                                    float* __restrict__ W2m) {
    const int n1 = T_S * IN_F * HID_F;
    const int n2 = T_S * HID_F * OUT_F;
    int i = blockIdx.x * 256 + threadIdx.x;
    if (i < n1) {
        W1m[i] = W1[i % (IN_F * HID_F)] * m1[i];
    } else if (i < n1 + n2) {
        int j = i - n1;
        W2m[j] = W2[j % (HID_F * OUT_F)] * m2[j];
    }
}

// -------------------------------------------------------- SpMM scatter-add
// one thread per (edge, 4-feature chunk); agg is L2-resident -> f32 atomics
__global__ __launch_bounds__(256) void spmm_scatter(const float* __restrict__ feat,
                                                    const float* __restrict__ ew,
                                                    const int*   __restrict__ src,
                                                    const int*   __restrict__ dst,
                                                    float*       __restrict__ agg,
                                                    int nedges, int D) {
    int idx = blockIdx.x * 256 + threadIdx.x;
    int chunks = D >> 2;
    int total = nedges * chunks;
    if (idx >= total) return;
    int e = idx / chunks;
    int q = (idx - e * chunks) << 2;
    float w = ew[e];
    const float4 v = *(const float4*)(feat + (size_t)src[e] * D + q);
    float* p = agg + (size_t)dst[e] * D + q;
    atomicAdd(p + 0, v.x * w);
    atomicAdd(p + 1, v.y * w);
    atomicAdd(p + 2, v.z * w);
    atomicAdd(p + 3, v.w * w);
}

// --------------------------------------------------- fp32 WMMA GEMM (+relu)
// D[M,Ncols] = A[M,K] @ W[K,Ncols] with V_WMMA_F32_16X16X4_F32.
// Each wave owns ONE 16-column N-tile: its B fragments (K/4 x v2f) are loaded
// once into registers and reused across a strided loop over 16-row M-tiles.
// Per M-tile: K/4 contiguous b64 A loads batched, then the unrolled WMMA chain.
template <int K, bool RELU>
__global__ __launch_bounds__(256) void gemm_wmma_f32(const float* __restrict__ A,
                                                     const float* __restrict__ W,
                                                     float*       __restrict__ Dm,
                                                     int M, int Ncols) {
    constexpr int KS = K / 4;                       // WMMA K-steps
    const int lane = threadIdx.x & 31;
    const int wid  = blockIdx.x * (blockDim.x >> 5) + (threadIdx.x >> 5);
    const int ntiles = Ncols >> 4;                  // 16-col N-tiles
    const int mtiles = M >> 4;
    const int nt = wid % ntiles;                    // this wave's N-tile
    const int wm = wid / ntiles;                    // position in M stride loop

    const int nbase = nt << 4;
    const int ml = lane & 15;                       // M (A) / N (B,C) lane index
    const int kh = lane >> 4;                       // half: K +2 (A,B), M +8 (C)

    // ---- B fragments: resident for the whole kernel ----
    v2f bfrag[KS];
#pragma unroll
    for (int ks = 0; ks < KS; ++ks) {
        const int ka = 4 * ks + 2 * kh;
        const float* w0 = W + (size_t)ka * Ncols + nbase + ml;
        bfrag[ks].x = w0[0];
        bfrag[ks].y = w0[Ncols];
    }

    // ---- strided loop over M-tiles, B reused from registers ----
    for (int mt = wm; mt < mtiles; mt += MWAVES) {
        const int mbase = mt << 4;
        const float* arow = A + (size_t)(mbase + ml) * K;

        v2f afrag[KS];
#pragma unroll
        for (int ks = 0; ks < KS; ++ks)             // batched b64 loads
            afrag[ks] = *(const v2f*)(arow + 4 * ks + 2 * kh);

        v8f acc = {};
#pragma unroll
        for (int ks = 0; ks < KS; ++ks)
            acc = __builtin_amdgcn_wmma_f32_16x16x4_f32(
                false, afrag[ks], false, bfrag[ks], (short)0, acc, false, false);

        // C/D layout: VGPR i -> row mbase + i + 8*kh, col nbase + ml
        float* dbase = Dm + (size_t)(mbase + 8 * kh) * Ncols + nbase + ml;
#pragma unroll
        for (int i = 0; i < 8; ++i) {
            float v = acc[i];
            if (RELU) v = fmaxf(v, 0.f);
            dbase[(size_t)i * Ncols] = v;
        }
    }
}

// ---------------------------------------------------------------- launcher
extern "C" void kernel_launch(void* const* d_in, const int* in_sizes, int n_in,
                              void* d_out, int out_size, void* d_ws, size_t ws_size,
                              hipStream_t stream) {
    const float* x   = (const float*)d_in[0];   // [N, IN]
    const float* ew  = (const float*)d_in[1];   // [E]
    const float* W1  = (const float*)d_in[2];   // [IN, HID]
    const float* W2  = (const float*)d_in[3];   // [HID, OUT]
    const float* m1  = (const float*)d_in[4];   // [T, IN, HID]
    const float* m2  = (const float*)d_in[5];   // [T, HID, OUT]
    const int*   src = (const int*)d_in[6];     // [E]
    const int*   dst = (const int*)d_in[7];     // [E]
    float*       out = (float*)d_out;           // [T, N, OUT]

    // workspace carve-out
    char* ws = (char*)d_ws;
    size_t off = 0;
    auto carve = [&](size_t bytes) -> float* {
        float* p = (float*)(ws + off);
        off = (off + bytes + 255) & ~(size_t)255;
        return p;
    };
    float* agg1 = carve((size_t)N_NODES * IN_F  * 4);
    float* h    = carve((size_t)N_NODES * HID_F * 4);
    float* agg2 = carve((size_t)N_NODES * HID_F * 4);
    float* W1m  = carve((size_t)T_S * IN_F  * HID_F * 4);
    float* W2m  = carve((size_t)T_S * HID_F * OUT_F * 4);
    (void)ws_size; (void)in_sizes; (void)n_in; (void)out_size;

    // 1. zero agg1, build masked weights
    {
        int n4 = (N_NODES * IN_F) / 4;
        zero_f32<<<(n4 + 255) / 256, 256, 0, stream>>>(agg1, n4);
        int nm = T_S * (IN_F * HID_F + HID_F * OUT_F);
        mask_weights<<<(nm + 255) / 256, 256, 0, stream>>>(W1, m1, W2, m2, W1m, W2m);
    }

    // 2. agg1 = scatter(x * w)   (mask-independent, shared across T)
    {
        int total = E_EDGES * (IN_F / 4);
        spmm_scatter<<<(total + 255) / 256, 256, 0, stream>>>(x, ew, src, dst, agg1,
                                                              E_EDGES, IN_F);
    }

    // 3. per-sample pipeline
    const int blocks1 = ((HID_F / 16) * MWAVES) / 8;   // gemm1: 8 N-tiles
    const int blocks2 = ((OUT_F / 16) * MWAVES) / 8;   // gemm2: 4 N-tiles
    const int n4h     = (N_NODES * HID_F) / 4;
    const int totalS2 = E_EDGES * (HID_F / 4);

    for (int t = 0; t < T_S; ++t) {
        // h = relu(agg1 @ W1m[t])
        gemm_wmma_f32<IN_F, true><<<blocks1, 256, 0, stream>>>(
            agg1, W1m + (size_t)t * IN_F * HID_F, h, N_NODES, HID_F);
        // agg2 = scatter(h * w)
        zero_f32<<<(n4h + 255) / 256, 256, 0, stream>>>(agg2, n4h);
        spmm_scatter<<<(totalS2 + 255) / 256, 256, 0, stream>>>(h, ew, src, dst, agg2,
                                                                E_EDGES, HID_F);
        // out[t] = agg2 @ W2m[t]
        gemm_wmma_f32<HID_F, false><<<blocks2, 256, 0, stream>>>(
            agg2, W2m + (size_t)t * HID_F * OUT_F,
            out + (size_t)t * N_NODES * OUT_F, N_NODES, OUT_F);
    }
}